// SimplifiedPhysicsDiffusionODE_22393959481652
// MI455X (gfx1250) — compile-verified
//
#include <hip/hip_runtime.h>
#include <math.h>

// ---------------------------------------------------------------------------
// SimplifiedPhysicsDiffusionODE for MI455X (gfx1250, wave32, WMMA).
//
// Strategy:
//  * zone_soft is numerically a hard one-hot -> 3-bit state. Precompute the
//    step MLP for all (t, cur) pairs: batch GEMMs (131072 x 256 x 128 x 64 x 8)
//    on the fp32 WMMA path (v_wmma_f32_16x16x4_f32).
//  * The argmax recurrence becomes a parallel scan over 8->8 transition maps
//    (packed as 8 nibbles in a u32; composition is associative).
//  * f_base is affine in norm_t -> reduced to a 256x8 table + 256-vec slope.
// ---------------------------------------------------------------------------

#define TT 16384
#define EPSF 1e-20f

typedef __attribute__((ext_vector_type(2))) float v2f;
typedef __attribute__((ext_vector_type(8))) float v8f;

__device__ __forceinline__ v8f wmma_f32_16x16x4(v2f a, v2f b, v8f c) {
  return __builtin_amdgcn_wmma_f32_16x16x4_f32(false, a, false, b, (short)0, c,
                                               false, false);
}

// ---------------------------------------------------------------------------
// Kernel 1: tiny precompute (tables, adjacency penalty, time normalization)
// ---------------------------------------------------------------------------
__global__ __launch_bounds__(256) void prep_kernel(
    const float* __restrict__ pa, const float* __restrict__ times,
    const int* __restrict__ ei, const float* __restrict__ W1,
    const float* __restrict__ b1, const float* __restrict__ D1,
    const float* __restrict__ d1, float* __restrict__ tbl,
    float* __restrict__ w1t, float* __restrict__ adjPen,
    float* __restrict__ aD, float* __restrict__ bD, float* __restrict__ scal) {
  __shared__ float Adj[64];
  int tid = threadIdx.x;
  if (tid < 64) Adj[tid] = 0.0f;
  __syncthreads();
  if (tid == 0) {
    for (int e = 0; e < 16; ++e) {
      int r = ei[e] & 7;
      int c = ei[16 + e] & 7;
      Adj[r * 8 + c] = 1.0f;
      Adj[c * 8 + r] = 1.0f;
    }
    for (int z = 0; z < 8; ++z) Adj[z * 8 + z] = 1.0f;
    float last = times[TT - 1];
    scal[0] = (last > 0.0f) ? last : 1.0f;  // denom
    scal[1] = (last > 0.0f) ? 1.0f : 0.0f;  // norm enable flag
  }
  __syncthreads();
  if (tid < 64) adjPen[tid] = -1000.0f * (1.0f - Adj[tid]);
  if (tid < 256) {
    const float* r = W1 + tid * 17;  // W1 is (256, 8+1+8) row-major
    float c1 = b1[tid];
    for (int d = 0; d < 8; ++d) c1 += r[d] * pa[d];
    w1t[tid] = r[8];
    for (int z = 0; z < 8; ++z) tbl[tid * 8 + z] = c1 + r[9 + z];
  }
  if (tid < 128) {
    const float* r = D1 + tid * 9;  // D1 is (128, 9)
    float a = d1[tid];
    for (int d = 0; d < 8; ++d) a += r[d] * pa[d];
    aD[tid] = a;
    bD[tid] = r[8];
  }
}

// ---------------------------------------------------------------------------
// Kernel 2: diversity MLP dv[t][0..7] = tanh(D3 relu(D2 relu(a + nt*b)))
// ---------------------------------------------------------------------------
__global__ __launch_bounds__(256) void diversity_kernel(
    const float* __restrict__ times, const float* __restrict__ D2,
    const float* __restrict__ d2, const float* __restrict__ D3,
    const float* __restrict__ d3, const float* __restrict__ aD,
    const float* __restrict__ bD, const float* __restrict__ scal,
    float* __restrict__ dvAll) {
  __shared__ float sD2[64 * 128];
  __shared__ float sD3[8 * 64];
  __shared__ float saD[128], sbD[128], sd2[64], sd3[8];
  int tid = threadIdx.x;
  for (int i = tid; i < 64 * 128; i += 256) sD2[i] = D2[i];
  for (int i = tid; i < 512; i += 256) sD3[i] = D3[i];
  if (tid < 128) {
    saD[tid] = aD[tid];
    sbD[tid] = bD[tid];
  }
  if (tid < 64) sd2[tid] = d2[tid];
  if (tid < 8) sd3[tid] = d3[tid];
  __syncthreads();

  int t = blockIdx.x * 256 + tid;
  float nt = (times[t] / scal[0]) * scal[1];
  float acc[64];
#pragma unroll
  for (int j = 0; j < 64; ++j) acc[j] = sd2[j];
#pragma unroll 4
  for (int i = 0; i < 128; ++i) {
    float h = fmaxf(saD[i] + nt * sbD[i], 0.0f);
#pragma unroll
    for (int j = 0; j < 64; ++j) acc[j] += sD2[j * 128 + i] * h;
  }
#pragma unroll
  for (int j = 0; j < 64; ++j) acc[j] = fmaxf(acc[j], 0.0f);
  for (int c = 0; c < 8; ++c) {
    float s = sd3[c];
#pragma unroll
    for (int j = 0; j < 64; ++j) s += sD3[c * 64 + j] * acc[j];
    dvAll[t * 8 + c] = tanhf(s);
  }
}

// ---------------------------------------------------------------------------
// Kernel 3: fused 3-GEMM WMMA pipeline. One wave per 16-row tile
// (rows = 2 timesteps x 8 cur-hypotheses). Bank-conflict-free LDS strides.
// ---------------------------------------------------------------------------
#define KS1 260  // 260 % 64 == 4 -> distinct banks across the 16 M-lanes
#define KS2 132
#define KS3 68

struct WaveBuf {
  float A[16 * KS1];  // H1, later H3, later score/nibble scratch
  float B[16 * KS2];  // H2
};

__global__ __launch_bounds__(64) void mlp_wmma_kernel(
    const float* __restrict__ times, const float* __restrict__ gum,
    const float* __restrict__ W2, const float* __restrict__ b2,
    const float* __restrict__ W3, const float* __restrict__ b3,
    const float* __restrict__ W4, const float* __restrict__ b4,
    const float* __restrict__ tbl, const float* __restrict__ w1t,
    const float* __restrict__ adjPen, const float* __restrict__ scal,
    const float* __restrict__ dvAll, float* __restrict__ conAll,
    unsigned* __restrict__ mapAll) {
  __shared__ WaveBuf buf[2];
  const int lane = threadIdx.x & 31;
  const int wv = threadIdx.x >> 5;
  const int lm = lane & 15;  // M (A row) or N (B/C col) lane id
  const int hl = lane >> 4;  // half-wave selector (K-pair / M+8)
  const int tile = blockIdx.x * 2 + wv;
  const int t0 = tile * 2;
  float* sA = buf[wv].A;
  float* sB = buf[wv].B;

  const float den = scal[0], flg = scal[1];
  const float nt0 = (times[t0] / den) * flg;
  const float nt1 = (times[t0 + 1] / den) * flg;

  // ---- H1[m][k] = relu(tbl[k][cur] + nt * w1t[k]) into LDS ----
  for (int idx = lane; idx < 16 * 256; idx += 32) {
    int m = idx >> 8;
    int k = idx & 255;
    float nt = (m < 8) ? nt0 : nt1;
    int cur = m & 7;
    float v = tbl[k * 8 + cur] + nt * w1t[k];
    sA[m * KS1 + k] = fmaxf(v, 0.0f);
  }
  __syncthreads();

  // ---- GEMM1: H2(16x128) = relu(H1 @ W2^T + b2), K=256 ----
  for (int n = 0; n < 8; ++n) {
    v8f acc = {0.f, 0.f, 0.f, 0.f, 0.f, 0.f, 0.f, 0.f};
    const float* wrow = W2 + (n * 16 + lm) * 256 + 2 * hl;
    const float* arow = sA + lm * KS1 + 2 * hl;
    for (int k0 = 0; k0 < 256; k0 += 4) {
      v2f a = *(const v2f*)(arow + k0);
      v2f b = *(const v2f*)(wrow + k0);
      acc = wmma_f32_16x16x4(a, b, acc);
    }
    float bc = b2[n * 16 + lm];
#pragma unroll
    for (int v = 0; v < 8; ++v)
      sB[(v + 8 * hl) * KS2 + n * 16 + lm] = fmaxf(acc[v] + bc, 0.0f);
  }
  __syncthreads();

  // ---- GEMM2: H3(16x64) = relu(H2 @ W3^T + b3), K=128 ----
  for (int n = 0; n < 4; ++n) {
    v8f acc = {0.f, 0.f, 0.f, 0.f, 0.f, 0.f, 0.f, 0.f};
    const float* wrow = W3 + (n * 16 + lm) * 128 + 2 * hl;
    const float* arow = sB + lm * KS2 + 2 * hl;
    for (int k0 = 0; k0 < 128; k0 += 4) {
      v2f a = *(const v2f*)(arow + k0);
      v2f b = *(const v2f*)(wrow + k0);
      acc = wmma_f32_16x16x4(a, b, acc);
    }
    float bc = b3[n * 16 + lm];
#pragma unroll
    for (int v = 0; v < 8; ++v)
      sA[(v + 8 * hl) * KS3 + n * 16 + lm] = fmaxf(acc[v] + bc, 0.0f);
  }
  __syncthreads();

  // ---- GEMM3: L(16x16, cols 8..15 are zero-padded W4) = H3 @ W4^T, K=64 ----
  v8f acc = {0.f, 0.f, 0.f, 0.f, 0.f, 0.f, 0.f, 0.f};
  {
    const float* arow = sA + lm * KS3 + 2 * hl;
    const float* wrow = W4 + lm * 64 + 2 * hl;  // valid only for lm < 8
    for (int k0 = 0; k0 < 64; k0 += 4) {
      v2f a = *(const v2f*)(arow + k0);
      v2f b = {0.0f, 0.0f};
      if (lm < 8) b = *(const v2f*)(wrow + k0);
      acc = wmma_f32_16x16x4(a, b, acc);
    }
  }

  // ---- Epilogue: constrained logits, gumbel scores, argmax -> packed map ----
  float* sScore = sA + 2048;      // past all GEMM3 reads (max idx 1083)
  float* sNib = sA + 2048 + 128;  // 16 nibbles as floats
  if (lm < 8) {
    int t = t0 + hl;
    int c = lm;
    float b4c = b4[c];
    float dvv = dvAll[t * 8 + c];
    float u = gum[t * 8 + c];
    float g = -logf(-logf(u + EPSF) + EPSF);
#pragma unroll
    for (int v = 0; v < 8; ++v) {  // v = hypothetical cur
      float val = acc[v] + b4c + 0.2f * dvv + adjPen[v * 8 + c];
      conAll[(t * 8 + v) * 8 + c] = val;
      sScore[hl * 64 + v * 8 + c] = val + g;
    }
  }
  __syncthreads();
  if (lane < 16) {
    int th = lane >> 3;
    int cur = lane & 7;
    const float* s = sScore + th * 64 + cur * 8;
    float best = s[0];
    int bi = 0;
#pragma unroll
    for (int j = 1; j < 8; ++j) {
      float x = s[j];
      if (x > best) { best = x; bi = j; }
    }
    sNib[lane] = (float)bi;
  }
  __syncthreads();
  if (lane < 2) {
    unsigned mp = 0;
    for (int i = 0; i < 8; ++i)
      mp |= ((unsigned)sNib[lane * 8 + i]) << (4 * i);
    mapAll[t0 + lane] = mp;
  }
}

// ---------------------------------------------------------------------------
// Kernel 4: parallel scan over 8->8 transition maps (nibble-packed u32).
// ---------------------------------------------------------------------------
#define IDMAP 0x76543210u

__device__ __forceinline__ unsigned comp8(unsigned g, unsigned f) {
  // (g o f)[i] = g[f[i]]   (f applied first)
  unsigned r = 0;
#pragma unroll
  for (int i = 0; i < 8; ++i) {
    unsigned fi = (f >> (4 * i)) & 7u;
    r |= ((g >> (4 * fi)) & 7u) << (4 * i);
  }
  return r;
}

__global__ __launch_bounds__(1024) void scan_kernel(
    const unsigned* __restrict__ mapAll, unsigned* __restrict__ zAll,
    float* __restrict__ outZones) {
  __shared__ unsigned s[1024];
  int tid = threadIdx.x;
  int base = tid * 16;  // 1024 threads x 16 maps = 16384
  unsigned c = IDMAP;
  for (int k = 0; k < 16; ++k) c = comp8(mapAll[base + k], c);
  s[tid] = c;
  __syncthreads();
  for (int off = 1; off < 1024; off <<= 1) {
    unsigned v = s[tid];
    if (tid >= off) v = comp8(v, s[tid - off]);
    __syncthreads();
    s[tid] = v;
    __syncthreads();
  }
  unsigned st = (tid == 0) ? IDMAP : s[tid - 1];  // exclusive prefix
  for (int k = 0; k < 16; ++k) {
    int t = base + k;
    zAll[t] = st & 7u;              // z_t (zone entering step t)
    st = comp8(mapAll[t], st);      // inclusive prefix
    outZones[t] = (float)(st & 7u); // traj_zones[t] = f_t(z_t)
  }
}

// ---------------------------------------------------------------------------
// Kernel 5: gather traj_logits[t][:] = constrained[t][z_t][:]
// ---------------------------------------------------------------------------
__global__ __launch_bounds__(256) void gather_kernel(
    const float* __restrict__ conAll, const unsigned* __restrict__ zAll,
    float* __restrict__ outLogits) {
  int idx = blockIdx.x * 256 + threadIdx.x;
  int t = idx >> 3;
  int j = idx & 7;
  unsigned z = zAll[t];
  outLogits[idx] = conAll[(t * 8u + z) * 8u + j];
}

// ---------------------------------------------------------------------------
extern "C" void kernel_launch(void* const* d_in, const int* in_sizes, int n_in,
                              void* d_out, int out_size, void* d_ws,
                              size_t ws_size, hipStream_t stream) {
  const float* pa = (const float*)d_in[0];
  const float* times = (const float*)d_in[1];
  const float* gum = (const float*)d_in[2];
  const int* ei = (const int*)d_in[3];
  const float* W1 = (const float*)d_in[4];
  const float* b1 = (const float*)d_in[5];
  const float* W2 = (const float*)d_in[6];
  const float* b2 = (const float*)d_in[7];
  const float* W3 = (const float*)d_in[8];
  const float* b3 = (const float*)d_in[9];
  const float* W4 = (const float*)d_in[10];
  const float* b4 = (const float*)d_in[11];
  const float* D1 = (const float*)d_in[12];
  const float* d1 = (const float*)d_in[13];
  const float* D2 = (const float*)d_in[14];
  const float* d2 = (const float*)d_in[15];
  const float* D3 = (const float*)d_in[16];
  const float* d3 = (const float*)d_in[17];
  // d_in[18] = tau: unused (outputs are pre-softmax; argmax is tau-invariant)
  (void)in_sizes; (void)n_in; (void)out_size; (void)ws_size;

  float* ws = (float*)d_ws;
  float* tbl = ws;             // 2048
  float* w1t = ws + 2048;      // 256
  float* adjPen = ws + 2304;   // 64
  float* aD = ws + 2368;       // 128
  float* bD = ws + 2496;       // 128
  float* scal = ws + 2624;     // 16
  float* dvAll = ws + 4096;    // T*8   = 131072
  float* conAll = ws + 135168; // T*8*8 = 1048576
  unsigned* mapAll = (unsigned*)(ws + 1183744);  // T u32
  unsigned* zAll = mapAll + TT;                  // T u32

  float* outLogits = (float*)d_out;
  float* outZones = outLogits + TT * 8;

  prep_kernel<<<1, 256, 0, stream>>>(pa, times, ei, W1, b1, D1, d1, tbl, w1t,
                                     adjPen, aD, bD, scal);
  diversity_kernel<<<TT / 256, 256, 0, stream>>>(times, D2, d2, D3, d3, aD, bD,
                                                 scal, dvAll);
  mlp_wmma_kernel<<<4096, 64, 0, stream>>>(times, gum, W2, b2, W3, b3, W4, b4,
                                           tbl, w1t, adjPen, scal, dvAll,
                                           conAll, mapAll);
  scan_kernel<<<1, 1024, 0, stream>>>(mapAll, zAll, outZones);
  gather_kernel<<<(TT * 8) / 256, 256, 0, stream>>>(conAll, zAll, outLogits);
}